// SeaCurrentRNN_3728031613159
// MI455X (gfx1250) — compile-verified
//
#include <hip/hip_runtime.h>
#include <hip/hip_bf16.h>

typedef __attribute__((ext_vector_type(2))) float v2f;
typedef __attribute__((ext_vector_type(8))) float v8f;

#define T_   16
#define H_   600
#define W_   600
#define HE   150
#define WE   150
#define C_   8
#define FLAT 180000   // 8*150*150
#define HID  256
#define SPLITS 180
#define KCHUNK 1000   // FLAT / SPLITS

// ---------------- workspace layout (float offsets), lifetime-aliased ----------------
// R_A: [0, 11.52M)          xe1 (K1->K2, 5.76M)  then y2 (K8->K9, 11.52M)
// R_B: [11.52M, 17.28M)     xe (2.88M, K2->K3a) + vp (0.72M, K3a->K3c) then y1 (5.76M, K7->K8)
// R_C: [17.28M, 23.04M)     c1 (5.76M, K3b->K3c) then partials(737280)+A(4096)+Hmat(4096)
// R_D: [23.04M, 25.92M)     featT (2.88M, K3c->K6)
// R_E: [25.92M, 28.80M)     outT  (2.88M, K6->K7)
#define OFF_XE1   ((size_t)0)
#define OFF_Y2    ((size_t)0)
#define OFF_XE    ((size_t)11520000)
#define OFF_VP    ((size_t)14400000)
#define OFF_Y1    ((size_t)11520000)
#define OFF_C1    ((size_t)17280000)
#define OFF_PART  ((size_t)17280000)
#define OFF_ABUF  ((size_t)18017280)
#define OFF_HMAT  ((size_t)18021376)
#define OFF_FEAT  ((size_t)23040000)
#define OFF_OUTT  ((size_t)25920000)
// total: 28,800,000 floats = 115.2 MB

// ------------------- K1: conv1(2->4,3x3,pad1)+relu+pool2 : 600^2 -> 300^2 -------------------
__global__ void k_enc1(const float* __restrict__ x, const float* __restrict__ w,
                       const float* __restrict__ b, float* __restrict__ y) {
  int i = blockIdx.x * blockDim.x + threadIdx.x;
  const int N = T_ * 4 * 300 * 300;
  if (i >= N) return;
  int pw = i % 300; int t2 = i / 300;
  int ph = t2 % 300; t2 /= 300;
  int o = t2 % 4; int t = t2 / 4;
  float s = 0.f;
  for (int dy = 0; dy < 2; ++dy)
    for (int dx = 0; dx < 2; ++dx) {
      int yy = 2 * ph + dy, xx = 2 * pw + dx;
      float acc = b[o];
      for (int c = 0; c < 2; ++c)
        for (int ky = 0; ky < 3; ++ky) {
          int sy = yy + ky - 1;
          if (sy < 0 || sy >= H_) continue;
          for (int kx = 0; kx < 3; ++kx) {
            int sx = xx + kx - 1;
            if (sx < 0 || sx >= W_) continue;
            acc += x[((size_t)(t * H_ + sy) * W_ + sx) * 2 + c] *
                   w[((o * 2 + c) * 3 + ky) * 3 + kx];
          }
        }
      s += fmaxf(acc, 0.f);
    }
  y[i] = 0.25f * s;
}

// ------------------- K2: conv2(4->8,3x3,pad1)+relu+pool2 : 300^2 -> 150^2 -------------------
__global__ void k_enc2(const float* __restrict__ xe1, const float* __restrict__ w,
                       const float* __restrict__ b, float* __restrict__ xe) {
  int i = blockIdx.x * blockDim.x + threadIdx.x;
  const int N = T_ * 8 * 150 * 150;
  if (i >= N) return;
  int pw = i % 150; int t2 = i / 150;
  int ph = t2 % 150; t2 /= 150;
  int o = t2 % 8; int t = t2 / 8;
  float s = 0.f;
  for (int dy = 0; dy < 2; ++dy)
    for (int dx = 0; dx < 2; ++dx) {
      int yy = 2 * ph + dy, xx = 2 * pw + dx;
      float acc = b[o];
      for (int c = 0; c < 4; ++c)
        for (int ky = 0; ky < 3; ++ky) {
          int sy = yy + ky - 1;
          if (sy < 0 || sy >= 300) continue;
          for (int kx = 0; kx < 3; ++kx) {
            int sx = xx + kx - 1;
            if (sx < 0 || sx >= 300) continue;
            acc += xe1[((size_t)(t * 4 + c) * 300 + sy) * 300 + sx] *
                   w[((o * 4 + c) * 3 + ky) * 3 + kx];
          }
        }
      s += fmaxf(acc, 0.f);
    }
  xe[i] = 0.25f * s;
}

// ------------------- K3a: vf = to_w@xe ; physics (periodic rolls) -> vp -------------------
__global__ void k_phys(const float* __restrict__ xe, const float* __restrict__ to_w,
                       const float* __restrict__ to_b, const float* __restrict__ cor_p,
                       const float* __restrict__ visc_p, const float* __restrict__ fric_p,
                       const float* __restrict__ dt_p, float* __restrict__ vp) {
  int i = blockIdx.x * blockDim.x + threadIdx.x;
  const int N = T_ * HE * WE;
  if (i >= N) return;
  int ww = i % WE; int t2 = i / WE;
  int h = t2 % HE; int t = t2 / HE;
  float cor = cor_p[0], visc = visc_p[0], fric = fric_p[0], dt = dt_p[0];
  const float* xb = xe + (size_t)t * C_ * HE * WE;
  auto vf = [&](int o, int hh, int wc) {
    float a = to_b[o];
    const float* p = xb + hh * WE + wc;
    #pragma unroll
    for (int c = 0; c < C_; ++c) a += to_w[o * C_ + c] * p[c * HE * WE];
    return a;
  };
  int hm = (h == 0) ? HE - 1 : h - 1, hp = (h == HE - 1) ? 0 : h + 1;
  int wm = (ww == 0) ? WE - 1 : ww - 1, wp = (ww == WE - 1) ? 0 : ww + 1;
  float u = vf(0, h, ww), v = vf(1, h, ww);
  float u_xm = vf(0, h, wm), u_xp = vf(0, h, wp), u_ym = vf(0, hm, ww), u_yp = vf(0, hp, ww);
  float v_xm = vf(1, h, wm), v_xp = vf(1, h, wp), v_ym = vf(1, hm, ww), v_yp = vf(1, hp, ww);
  float du_dx = (u_xp - u_xm) * 0.5f, du_dy = (u_yp - u_ym) * 0.5f;
  float dv_dx = (v_xp - v_xm) * 0.5f, dv_dy = (v_yp - v_ym) * 0.5f;
  float lap_u = u_xm + u_xp + u_ym + u_yp - 4.f * u;
  float lap_v = v_xm + v_xp + v_ym + v_yp - 4.f * v;
  float du = -(u * du_dx + v * du_dy) + cor * v + visc * lap_u - fric * u;
  float dv = -(u * dv_dx + v * dv_dy) - cor * u + visc * lap_v - fric * v;
  vp[((size_t)t * 2 + 0) * HE * WE + h * WE + ww] = u + dt * du;
  vp[((size_t)t * 2 + 1) * HE * WE + h * WE + ww] = v + dt * dv;
}

// ------------------- K3b: corr1 conv (2->16,3x3,pad1) + relu -------------------
__global__ void k_corr1(const float* __restrict__ vp, const float* __restrict__ w,
                        const float* __restrict__ b, float* __restrict__ c1) {
  int i = blockIdx.x * blockDim.x + threadIdx.x;
  const int N = T_ * 16 * HE * WE;
  if (i >= N) return;
  int ww = i % WE; int t2 = i / WE;
  int h = t2 % HE; t2 /= HE;
  int o = t2 % 16; int t = t2 / 16;
  float acc = b[o];
  for (int c = 0; c < 2; ++c)
    for (int ky = 0; ky < 3; ++ky) {
      int sy = h + ky - 1;
      if (sy < 0 || sy >= HE) continue;
      for (int kx = 0; kx < 3; ++kx) {
        int sx = ww + kx - 1;
        if (sx < 0 || sx >= WE) continue;
        acc += vp[((size_t)(t * 2 + c) * HE + sy) * WE + sx] *
               w[((o * 2 + c) * 3 + ky) * 3 + kx];
      }
    }
  c1[i] = fmaxf(acc, 0.f);
}

// ------------------- K3c: corr2 conv (16->2), updated = vp+(1-s)*corr, feat = from_w@updated
//                          store feat TRANSPOSED: featT[k*16 + t] -------------------
__global__ void k_corr2_feat(const float* __restrict__ c1, const float* __restrict__ vp,
                             const float* __restrict__ w2, const float* __restrict__ b2,
                             const float* __restrict__ from_w, const float* __restrict__ from_b,
                             const float* __restrict__ pw_p, float* __restrict__ featT) {
  int i = blockIdx.x * blockDim.x + threadIdx.x;
  const int N = T_ * HE * WE;
  if (i >= N) return;
  int ww = i % WE; int t2 = i / WE;
  int h = t2 % HE; int t = t2 / HE;
  float s = 1.f / (1.f + expf(-pw_p[0]));
  float upd[2];
  for (int o = 0; o < 2; ++o) {
    float acc = b2[o];
    for (int c = 0; c < 16; ++c)
      for (int ky = 0; ky < 3; ++ky) {
        int sy = h + ky - 1;
        if (sy < 0 || sy >= HE) continue;
        for (int kx = 0; kx < 3; ++kx) {
          int sx = ww + kx - 1;
          if (sx < 0 || sx >= WE) continue;
          acc += c1[((size_t)(t * 16 + c) * HE + sy) * WE + sx] *
                 w2[((o * 16 + c) * 3 + ky) * 3 + kx];
        }
      }
    float vpv = vp[((size_t)t * 2 + o) * HE * WE + h * WE + ww];
    upd[o] = vpv + (1.f - s) * acc;
  }
  #pragma unroll
  for (int c = 0; c < C_; ++c) {
    float f = from_b[c] + from_w[c * 2 + 0] * upd[0] + from_w[c * 2 + 1] * upd[1];
    size_t k = (size_t)c * (HE * WE) + h * WE + ww;
    featT[k * T_ + t] = f;
  }
}

// ------------------- K4: GEMM1 split-K  A(256x16) = W_in(256xFLAT) @ featT(FLATx16) ----
// one wave per (16-row tile, K-split); V_WMMA_F32_16X16X4_F32 K-loop
__global__ void k_gemm1(const float* __restrict__ Win, const float* __restrict__ featT,
                        float* __restrict__ partials) {
  int lane = threadIdx.x;                 // 0..31, no divergence (EXEC all ones)
  int mtile = blockIdx.x;                 // 0..15
  int s = blockIdx.y;                     // 0..SPLITS-1
  int kbase = s * KCHUNK;
  int n = lane & 15;
  int khalf = (lane >> 4) * 2;            // 0 or 2
  const float* Ap = Win + (size_t)(mtile * 16 + n) * FLAT + kbase + khalf;
  const float* Bp = featT + (size_t)(kbase + khalf) * T_ + n;
  v8f acc = {0.f, 0.f, 0.f, 0.f, 0.f, 0.f, 0.f, 0.f};
  for (int k = 0; k < KCHUNK; k += 4) {
    v2f a = *(const v2f*)(Ap + k);        // A[M=n][K=khalf,khalf+1], 8B aligned
    v2f b;
    b.x = Bp[(size_t)k * T_];             // B[K=khalf  ][N=n]
    b.y = Bp[(size_t)(k + 1) * T_];       // B[K=khalf+1][N=n]
    acc = __builtin_amdgcn_wmma_f32_16x16x4_f32(false, a, false, b, (short)0, acc,
                                                false, false);
  }
  float* P = partials + (size_t)s * (HID * 16);
  int mo = mtile * 16 + (lane >> 4) * 8;
  #pragma unroll
  for (int r = 0; r < 8; ++r) P[(mo + r) * 16 + n] = acc[r];
}

// ------------------- K4b: deterministic split-K reduce + W_in_b -------------------
__global__ void k_reduce(const float* __restrict__ partials, const float* __restrict__ Win_b,
                         float* __restrict__ A) {
  int i = blockIdx.x * blockDim.x + threadIdx.x;
  if (i >= HID * 16) return;
  float acc = Win_b[i >> 4];
  for (int s = 0; s < SPLITS; ++s) acc += partials[(size_t)s * HID * 16 + i];
  A[i] = acc;
}

// ------------------- K5: sequential recurrence (tiny): h_t = tanh(A[:,t] + Whid@h + b) ----
__global__ void k_rnn(const float* __restrict__ A, const float* __restrict__ Whid,
                      const float* __restrict__ Whid_b, float* __restrict__ Hmat) {
  __shared__ float h[HID];
  int r = threadIdx.x;
  h[r] = 0.f;
  __syncthreads();
  for (int t = 0; t < T_; ++t) {
    float acc = A[r * 16 + t] + Whid_b[r];
    const float* Wr = Whid + (size_t)r * HID;
    for (int j = 0; j < HID; ++j) acc += Wr[j] * h[j];
    float v = tanhf(acc);
    __syncthreads();
    h[r] = v;
    Hmat[r * 16 + t] = v;                 // B-matrix layout for GEMM2: Hmat[k*16 + t]
    __syncthreads();
  }
}

// ------------------- K6: GEMM2 + epilogue: outT = featT + 0.1*(W_out @ H + b) ----
// one wave per 16-row tile of M=FLAT; full K=256 in-wave
__global__ void k_gemm2(const float* __restrict__ Wout, const float* __restrict__ Hmat,
                        const float* __restrict__ Wout_b, const float* __restrict__ featT,
                        float* __restrict__ outT) {
  int lane = threadIdx.x;
  size_t f0 = (size_t)blockIdx.x * 16;
  int n = lane & 15;
  int khalf = (lane >> 4) * 2;
  const float* Ap = Wout + (f0 + n) * HID + khalf;
  const float* Bp = Hmat + khalf * 16 + n;
  v8f acc = {0.f, 0.f, 0.f, 0.f, 0.f, 0.f, 0.f, 0.f};
  #pragma unroll 4
  for (int k = 0; k < HID; k += 4) {
    v2f a = *(const v2f*)(Ap + k);
    v2f b;
    b.x = Bp[k * 16];
    b.y = Bp[(k + 1) * 16];
    acc = __builtin_amdgcn_wmma_f32_16x16x4_f32(false, a, false, b, (short)0, acc,
                                                false, false);
  }
  size_t fo = f0 + (size_t)((lane >> 4) * 8);
  #pragma unroll
  for (int r = 0; r < 8; ++r) {
    size_t f = fo + r;
    outT[f * 16 + n] = featT[f * 16 + n] + 0.1f * (acc[r] + Wout_b[f]);
  }
}

// ------------------- K7: deconv1 (8->4, 2x2) + relu : 150^2 -> 300^2 -------------------
__global__ void k_dec1(const float* __restrict__ outT, const float* __restrict__ w,
                       const float* __restrict__ b, float* __restrict__ y1) {
  int i = blockIdx.x * blockDim.x + threadIdx.x;
  const int N = T_ * 4 * 300 * 300;
  if (i >= N) return;
  int X = i % 300; int t2 = i / 300;
  int Y = t2 % 300; t2 /= 300;
  int o = t2 % 4; int t = t2 / 4;
  int h = Y >> 1, p = Y & 1, ww = X >> 1, q = X & 1;
  float acc = b[o];
  #pragma unroll
  for (int c = 0; c < 8; ++c)
    acc += outT[((size_t)c * HE * WE + h * WE + ww) * T_ + t] *
           w[((c * 4 + o) * 2 + p) * 2 + q];
  y1[i] = fmaxf(acc, 0.f);
}

// ------------------- K8: deconv2 (4->2, 2x2) + relu : 300^2 -> 600^2 -------------------
__global__ void k_dec2(const float* __restrict__ y1, const float* __restrict__ w,
                       const float* __restrict__ b, float* __restrict__ y2) {
  int i = blockIdx.x * blockDim.x + threadIdx.x;
  const int N = T_ * 2 * 600 * 600;
  if (i >= N) return;
  int X = i % 600; int t2 = i / 600;
  int Y = t2 % 600; t2 /= 600;
  int o = t2 % 2; int t = t2 / 2;
  int h = Y >> 1, p = Y & 1, ww = X >> 1, q = X & 1;
  float acc = b[o];
  #pragma unroll
  for (int c = 0; c < 4; ++c)
    acc += y1[((size_t)(t * 4 + c) * 300 + h) * 300 + ww] *
           w[((c * 2 + o) * 2 + p) * 2 + q];
  y2[i] = fmaxf(acc, 0.f);
}

// ------------------- K9: convf (2->2,3x3,pad1), write NHWC output -------------------
__global__ void k_final(const float* __restrict__ y2, const float* __restrict__ w,
                        const float* __restrict__ b, float* __restrict__ out) {
  int i = blockIdx.x * blockDim.x + threadIdx.x;
  const int N = T_ * H_ * W_;
  if (i >= N) return;
  int x = i % W_; int t2 = i / W_;
  int h = t2 % H_; int t = t2 / H_;
  #pragma unroll
  for (int o = 0; o < 2; ++o) {
    float acc = b[o];
    for (int c = 0; c < 2; ++c)
      for (int ky = 0; ky < 3; ++ky) {
        int sy = h + ky - 1;
        if (sy < 0 || sy >= H_) continue;
        for (int kx = 0; kx < 3; ++kx) {
          int sx = x + kx - 1;
          if (sx < 0 || sx >= W_) continue;
          acc += y2[((size_t)(t * 2 + c) * H_ + sy) * W_ + sx] *
                 w[((o * 2 + c) * 3 + ky) * 3 + kx];
        }
      }
    out[(size_t)i * 2 + o] = acc;
  }
}

extern "C" void kernel_launch(void* const* d_in, const int* in_sizes, int n_in,
                              void* d_out, int out_size, void* d_ws, size_t ws_size,
                              hipStream_t stream) {
  const float* x        = (const float*)d_in[0];
  const float* conv1_w  = (const float*)d_in[1];
  const float* conv1_b  = (const float*)d_in[2];
  const float* conv2_w  = (const float*)d_in[3];
  const float* conv2_b  = (const float*)d_in[4];
  const float* to_w     = (const float*)d_in[5];
  const float* to_b     = (const float*)d_in[6];
  const float* coriolis = (const float*)d_in[7];
  const float* viscosity= (const float*)d_in[8];
  const float* friction = (const float*)d_in[9];
  const float* dt       = (const float*)d_in[10];
  const float* phys_w   = (const float*)d_in[11];
  const float* corr1_w  = (const float*)d_in[12];
  const float* corr1_b  = (const float*)d_in[13];
  const float* corr2_w  = (const float*)d_in[14];
  const float* corr2_b  = (const float*)d_in[15];
  const float* from_w   = (const float*)d_in[16];
  const float* from_b   = (const float*)d_in[17];
  const float* W_in_w   = (const float*)d_in[18];
  const float* W_in_b   = (const float*)d_in[19];
  const float* W_hid_w  = (const float*)d_in[20];
  const float* W_hid_b  = (const float*)d_in[21];
  const float* W_out_w  = (const float*)d_in[22];
  const float* W_out_b  = (const float*)d_in[23];
  const float* dec1_w   = (const float*)d_in[24];
  const float* dec1_b   = (const float*)d_in[25];
  const float* dec2_w   = (const float*)d_in[26];
  const float* dec2_b   = (const float*)d_in[27];
  const float* convf_w  = (const float*)d_in[28];
  const float* convf_b  = (const float*)d_in[29];

  float* ws    = (float*)d_ws;
  float* xe1   = ws + OFF_XE1;
  float* xe    = ws + OFF_XE;
  float* vp    = ws + OFF_VP;
  float* c1    = ws + OFF_C1;
  float* featT = ws + OFF_FEAT;
  float* part  = ws + OFF_PART;
  float* Abuf  = ws + OFF_ABUF;
  float* Hmat  = ws + OFF_HMAT;
  float* outT  = ws + OFF_OUTT;
  float* y1    = ws + OFF_Y1;
  float* y2    = ws + OFF_Y2;
  float* out   = (float*)d_out;

  const int B = 256;
  // encoder
  k_enc1<<<dim3((T_ * 4 * 300 * 300 + B - 1) / B), B, 0, stream>>>(x, conv1_w, conv1_b, xe1);
  k_enc2<<<dim3((T_ * 8 * 150 * 150 + B - 1) / B), B, 0, stream>>>(xe1, conv2_w, conv2_b, xe);
  // per-frame physics + correction + feature projection (h-independent -> fully parallel)
  k_phys<<<dim3((T_ * HE * WE + B - 1) / B), B, 0, stream>>>(
      xe, to_w, to_b, coriolis, viscosity, friction, dt, vp);
  k_corr1<<<dim3((T_ * 16 * HE * WE + B - 1) / B), B, 0, stream>>>(vp, corr1_w, corr1_b, c1);
  k_corr2_feat<<<dim3((T_ * HE * WE + B - 1) / B), B, 0, stream>>>(
      c1, vp, corr2_w, corr2_b, from_w, from_b, phys_w, featT);
  // batched input-projection GEMM (WMMA, split-K) + reduce
  k_gemm1<<<dim3(16, SPLITS), 32, 0, stream>>>(W_in_w, featT, part);
  k_reduce<<<dim3((HID * 16 + B - 1) / B), B, 0, stream>>>(part, W_in_b, Abuf);
  // tiny sequential recurrence
  k_rnn<<<1, HID, 0, stream>>>(Abuf, W_hid_w, W_hid_b, Hmat);
  // batched output-projection GEMM (WMMA) fused with epilogue
  k_gemm2<<<dim3(FLAT / 16), 32, 0, stream>>>(W_out_w, Hmat, W_out_b, featT, outT);
  // decoder
  k_dec1<<<dim3((T_ * 4 * 300 * 300 + B - 1) / B), B, 0, stream>>>(outT, dec1_w, dec1_b, y1);
  k_dec2<<<dim3((T_ * 2 * 600 * 600 + B - 1) / B), B, 0, stream>>>(y1, dec2_w, dec2_b, y2);
  k_final<<<dim3((T_ * H_ * W_ + B - 1) / B), B, 0, stream>>>(y2, convf_w, convf_b, out);
}